// SampleScoreModel_88373247082964
// MI455X (gfx1250) — compile-verified
//
#include <hip/hip_runtime.h>
#include <hip/hip_bf16.h>

typedef __attribute__((ext_vector_type(2))) float v2f;
typedef __attribute__((ext_vector_type(8))) float v8f;

#define NSAMP 65536
#define NBATCH 32
#define DIM 64
#define SLOT_FLOATS 1056   // 16 m + 16 s + 16*64 U
#define MAX_BLOCKS 128

// ---------------------------------------------------------------------------
// Kernel 1: fused  logits-GEMM + online softmax + weighted-sum-GEMM partials
//
//  logit[n][b] = (s_n . (x_b/denom_b)) - 0.5*||s_n||^2/denom_b - 0.5*||x_b||^2/denom_b
//  The two bias terms ride in an extra K-step of the same WMMA chain.
//  GEMM2 is computed transposed (C[M=d][N=batch]) so the online-softmax
//  rescale factor is a per-lane scalar (no cross-lane broadcasts).
// ---------------------------------------------------------------------------
__global__ __launch_bounds__(256) void score_partials_kernel(
    const float* __restrict__ t, const float* __restrict__ x,
    const float* __restrict__ samples, float* __restrict__ ws, int G)
{
    __shared__ float xs[NBATCH * DIM];       // x/denom, row-major [32][64]
    __shared__ float invd_s[NBATCH];         // 1/denom per batch
    __shared__ float xnh_s[NBATCH];          // 0.5*||x_b||^2
    __shared__ float smp[8][16 * DIM];       // per-wave 16x64 sample tile
    __shared__ float etile[8][16 * 16];      // per-wave e-tile (sample-major)

    const int tid  = threadIdx.x;
    const int wave = tid >> 5;
    const int lane = tid & 31;
    const int lo   = lane & 15;   // 0..15
    const int hi   = lane >> 4;   // 0 or 1
    const int sw   = wave >> 1;   // sample subtile 0..3
    const int bw   = wave & 1;    // batch half 0..1

    // ---- stage x and per-batch constants ----
    for (int i = tid; i < NBATCH * DIM; i += 256) xs[i] = x[i];
    if (tid < NBATCH) {
        // sigma(t) = 0.01 * 10000^t = 0.01 * exp(t*ln(10000))
        float ts  = 0.01f * __expf(t[tid] * 9.210340371976184f);
        float den = ts * ts;
        invd_s[tid] = 1.0f / den;
        float xn = 0.0f;
        for (int k = 0; k < DIM; ++k) { float v = x[tid * DIM + k]; xn += v * v; }
        xnh_s[tid] = 0.5f * xn;
    }
    __syncthreads();
    // pre-scale x rows by 1/denom so GEMM1 directly produces logits
    for (int i = tid; i < NBATCH * DIM; i += 256) xs[i] *= invd_s[i >> 6];
    __syncthreads();

    const int   myb  = bw * 16 + lo;     // batch handled by this lane's column
    const float invd = invd_s[myb];
    const float xnh  = xnh_s[myb];

    // bias fragment for the extra K-step of GEMM1 (loop-invariant):
    //   B[64][b] = -0.5/denom_b ; B[65][b] = -0.5*||x_b||^2/denom_b ; K=66,67 pad = 0
    v2f bfold;
    bfold[0] = hi ? 0.0f : (-0.5f * invd);
    bfold[1] = hi ? 0.0f : (-xnh * invd);

    float m_run = -3.0e38f;
    float s_run = 0.0f;
    v8f acc2[4];                          // U tile (transposed): 64 d x 16 batches
    #pragma unroll
    for (int dt = 0; dt < 4; ++dt)
        #pragma unroll
        for (int r = 0; r < 8; ++r) acc2[dt][r] = 0.0f;

    float* mysmp = smp[wave];
    float* mye   = etile[wave];

    const int nchunks = NSAMP / 64;       // 1024 chunks of 64 samples
    for (int c = blockIdx.x; c < nchunks; c += G) {
        const int n0 = c * 64 + sw * 16;  // this wave's 16-sample rows

        // prefetch next chunk while we work on this one
        if (c + G < nchunks)
            __builtin_prefetch(samples + (size_t)(c + G) * 64 * DIM + tid * 16, 0, 0);

        // ---- stage 16x64 samples into wave-private LDS; fold ||s||^2 ----
        float sq = 0.0f;
        {
            const float* src = samples + (size_t)(n0 + lo) * DIM + hi * 32;
            float*       dst = mysmp + lo * DIM + hi * 32;
            #pragma unroll
            for (int i = 0; i < 8; ++i) {
                float4 v = ((const float4*)src)[i];
                ((float4*)dst)[i] = v;
                sq += v.x * v.x + v.y * v.y + v.z * v.z + v.w * v.w;
            }
        }
        sq += __shfl_xor(sq, 16, 32);     // all lanes: ||s||^2 of row (lane&15)

        // ---- GEMM1: logits tile via f32 WMMA, K=64 dot + 1 bias step ----
        v8f accl0, accl1;
        #pragma unroll
        for (int r = 0; r < 8; ++r) { accl0[r] = 0.0f; accl1[r] = 0.0f; }
        #pragma unroll
        for (int ks = 0; ks < 16; ks += 2) {
            int k0 = ks * 4 + 2 * hi;
            v2f a0 = *(const v2f*)(mysmp + lo * DIM + k0);
            v2f b0 = *(const v2f*)(xs + myb * DIM + k0);
            accl0 = __builtin_amdgcn_wmma_f32_16x16x4_f32(
                        false, a0, false, b0, (short)0, accl0, false, false);
            int k1 = (ks + 1) * 4 + 2 * hi;
            v2f a1 = *(const v2f*)(mysmp + lo * DIM + k1);
            v2f b1 = *(const v2f*)(xs + myb * DIM + k1);
            accl1 = __builtin_amdgcn_wmma_f32_16x16x4_f32(
                        false, a1, false, b1, (short)0, accl1, false, false);
        }
        {   // bias K-step: A[:,64]=||s||^2, A[:,65]=1 (hi-lane pad rows hit zero B)
            v2f afold; afold[0] = sq; afold[1] = 1.0f;
            accl1 = __builtin_amdgcn_wmma_f32_16x16x4_f32(
                        false, afold, false, bfold, (short)0, accl1, false, false);
        }

        // ---- logits are finished in the accumulator; tile max per batch ----
        float l[8];
        float tm = -3.0e38f;
        #pragma unroll
        for (int r = 0; r < 8; ++r) {
            l[r] = accl0[r] + accl1[r];
            tm = fmaxf(tm, l[r]);
        }
        tm = fmaxf(tm, __shfl_xor(tm, 16, 32));          // column max per batch
        float m_new = fmaxf(m_run, tm);
        float sc    = __expf(m_run - m_new);             // first iter: exp(-inf)=0

        float e[8], tsum = 0.0f;
        #pragma unroll
        for (int r = 0; r < 8; ++r) { e[r] = __expf(l[r] - m_new); tsum += e[r]; }
        tsum += __shfl_xor(tsum, 16, 32);
        s_run = s_run * sc + tsum;
        m_run = m_new;

        // ---- rescale U accumulators: batch == lane column, factor is scalar ----
        #pragma unroll
        for (int dt = 0; dt < 4; ++dt)
            #pragma unroll
            for (int r = 0; r < 8; ++r) acc2[dt][r] *= sc;

        // ---- e-tile to LDS, sample-major: e[(sample)*16 + batch] ----
        #pragma unroll
        for (int r = 0; r < 8; ++r) mye[(r + 8 * hi) * 16 + lo] = e[r];

        // ---- GEMM2 (transposed): U[d][b] += samples^T[64x16] x e[16x16] ----
        #pragma unroll
        for (int ks = 0; ks < 4; ++ks) {
            int k0 = ks * 4 + 2 * hi;
            v2f b; b[0] = mye[k0 * 16 + lo]; b[1] = mye[(k0 + 1) * 16 + lo];
            #pragma unroll
            for (int dt = 0; dt < 4; ++dt) {
                v2f a;
                a[0] = mysmp[k0 * DIM + dt * 16 + lo];
                a[1] = mysmp[(k0 + 1) * DIM + dt * 16 + lo];
                acc2[dt] = __builtin_amdgcn_wmma_f32_16x16x4_f32(
                               false, a, false, b, (short)0, acc2[dt], false, false);
            }
        }
    }

    // ---- write per-wave partial: m[16], s[16], U[16 batches][64 d] ----
    float* slot = ws + (size_t)(blockIdx.x * 8 + wave) * SLOT_FLOATS;
    if (hi == 0) { slot[lo] = m_run; slot[16 + lo] = s_run; }
    // transposed C layout: row d = dt*16 + r + 8*hi, col batch = lo
    #pragma unroll
    for (int dt = 0; dt < 4; ++dt)
        #pragma unroll
        for (int r = 0; r < 8; ++r)
            slot[32 + lo * DIM + dt * 16 + r + 8 * hi] = acc2[dt][r];
}

// ---------------------------------------------------------------------------
// Kernel 2: log-sum-exp merge of partials + final scale
// ---------------------------------------------------------------------------
__global__ __launch_bounds__(64) void score_finalize_kernel(
    const float* __restrict__ t, const float* __restrict__ x,
    const float* __restrict__ ws, float* __restrict__ out, int G)
{
    const int b   = blockIdx.x;     // 0..31
    const int d   = threadIdx.x;    // 0..63
    const int bwh = b >> 4;         // which batch half
    const int bb  = b & 15;

    float m = -3.0e38f;
    for (int g = 0; g < G; ++g)
        for (int j = 0; j < 4; ++j) {
            const float* slot = ws + (size_t)(g * 8 + 2 * j + bwh) * SLOT_FLOATS;
            m = fmaxf(m, slot[bb]);
        }
    float S = 0.0f, U = 0.0f;
    for (int g = 0; g < G; ++g)
        for (int j = 0; j < 4; ++j) {
            const float* slot = ws + (size_t)(g * 8 + 2 * j + bwh) * SLOT_FLOATS;
            float w = __expf(slot[bb] - m);
            S += slot[16 + bb] * w;
            U += slot[32 + bb * DIM + d] * w;
        }
    float ts  = 0.01f * __expf(t[b] * 9.210340371976184f);
    float den = ts * ts;
    out[b * DIM + d] = (ts / den) * (U / S - x[b * DIM + d]);
}

// ---------------------------------------------------------------------------
extern "C" void kernel_launch(void* const* d_in, const int* in_sizes, int n_in,
                              void* d_out, int out_size, void* d_ws, size_t ws_size,
                              hipStream_t stream) {
    const float* t       = (const float*)d_in[0];  // [32]
    const float* x       = (const float*)d_in[1];  // [32,64]
    const float* samples = (const float*)d_in[2];  // [65536,64]
    float*       out     = (float*)d_out;          // [32,64]
    float*       ws      = (float*)d_ws;

    const size_t per_block = (size_t)8 * SLOT_FLOATS * sizeof(float);
    int G = (int)(ws_size / per_block);
    if (G < 1) G = 1;
    if (G > MAX_BLOCKS) G = MAX_BLOCKS;

    score_partials_kernel<<<G, 256, 0, stream>>>(t, x, samples, ws, G);
    score_finalize_kernel<<<NBATCH, DIM, 0, stream>>>(t, x, ws, out, G);
}